// QRNN_54511724921108
// MI455X (gfx1250) — compile-verified
//
#include <hip/hip_runtime.h>
#include <math.h>

typedef __attribute__((ext_vector_type(2))) float v2f;
typedef __attribute__((ext_vector_type(4))) float v4f;
typedef __attribute__((ext_vector_type(8))) float v8f;
typedef __attribute__((ext_vector_type(4))) unsigned int u32x4;
typedef __attribute__((ext_vector_type(4))) int i32x4;
typedef __attribute__((ext_vector_type(8))) int i32x8;

#define B_    8
#define T_    4096
#define C_    512
#define U_    512
#define NG    1536          // 3*U
#define KTOT  1024          // 2*C
#define ROWS  (B_ * T_)     // 32768
#define NELEM ((size_t)ROWS * C_)

#define BLK_M 64
#define BLK_N 128
#define KC    64
#define NCHUNK (KTOT / KC)  // 16
#define APAD  66            // 64 k + 2 pad  -> conflict-free, b64-aligned
#define BPAD  66

#if defined(__HIP_DEVICE_COMPILE__) && __has_builtin(__builtin_amdgcn_tensor_load_to_lds) && __has_builtin(__builtin_amdgcn_s_wait_tensorcnt)
#define USE_TDM 1
#else
#define USE_TDM 0
#endif

#if USE_TDM
// 2D tile DMA global -> LDS via Tensor Data Mover. Dim0 = contiguous.
__device__ __forceinline__ void tdm_tile_load(const float* gsrc, unsigned ldsoff,
                                              int tdim0, int tdim1, int til0, int til1,
                                              int stride0, unsigned flags)
{
  unsigned long long ga = (unsigned long long)(uintptr_t)gsrc;
  u32x4 g0;
  g0[0] = 1u;                                              // count=1, user mode
  g0[1] = ldsoff;                                          // lds_addr (bytes)
  g0[2] = (unsigned)(ga & 0xffffffffu);                    // global_addr[31:0]
  g0[3] = (unsigned)((ga >> 32) & 0x1ffffffu) | (2u << 30);// addr[56:32] | type=2
  i32x8 g1;
  g1[0] = (int)flags;                                      // data_size / pad fields
  g1[1] = (int)((tdim0 & 0xffff) << 16);                   // tensor_dim0[15:0]
  g1[2] = (int)(((tdim0 >> 16) & 0xffff) | ((tdim1 & 0xffff) << 16));
  g1[3] = (int)(((tdim1 >> 16) & 0xffff) | (til0 << 16));  // tile_dim0
  g1[4] = til1;                                            // tile_dim1 | tile_dim2=0
  g1[5] = stride0;                                         // tensor_dim0_stride
  g1[6] = 0;
  g1[7] = 0;
  i32x4 gz = (i32x4){0, 0, 0, 0};
#if __clang_major__ >= 23
  i32x8 gz8 = (i32x8){0, 0, 0, 0, 0, 0, 0, 0};
  __builtin_amdgcn_tensor_load_to_lds(g0, g1, gz, gz, gz8, 0);
#else
  __builtin_amdgcn_tensor_load_to_lds(g0, g1, gz, gz, 0);
#endif
}
// data_size=4B | pad_enable | pad_interval=64 dwords (code 5) | pad_amount=2 dwords (code 1)
#define TDM_FLAGS_A ((2u << 16) | (1u << 20) | (5u << 22) | (1u << 25))
#endif

// gates = X2 @ K + bias, fused activations.
// X2[r, 0:1024] == flat inputs at offset r*512 (rows overlap in memory), with
// the upper 512 k-values zeroed when (r % T) == T-1 (batch edge).
__global__ __launch_bounds__(256) void qrnn_gemm_gates(
    const float* __restrict__ in,    // [B*T*C] flat
    const float* __restrict__ kw,    // [KTOT, NG] row-major
    const float* __restrict__ bias,  // [NG]
    float* __restrict__ xact, float* __restrict__ fgate, float* __restrict__ ogate)
{
  __shared__ float ldsA[2][BLK_M * APAD];   // [row][k], padded, 2 x 16.9 KB
  __shared__ float ldsB[2][BLK_N * BPAD];   // [n][k] transposed, 2 x 33.8 KB

  const int tid     = threadIdx.x;
  const int lane    = tid & 31;
  const int wave    = tid >> 5;
  const int waveM   = wave >> 1;                 // 0..3
  const int waveN   = wave & 1;                  // 0..1

  const int r0      = blockIdx.y * BLK_M;
  const int m0      = r0 + waveM * 16;
  const int nblk0   = blockIdx.x * BLK_N;
  const int nwave   = waveN * 64;

  const int  halfrow = lane & 15;
  const int  sel2    = (lane >> 4) << 1;         // 0 (lanes 0-15) or 2 (16-31)
  const int  r       = m0 + halfrow;             // A-matrix global row
  const bool lastT   = ((r & (T_ - 1)) == (T_ - 1));

  v8f acc[4];
  #pragma unroll
  for (int i = 0; i < 4; ++i) acc[i] = (v8f){0.f,0.f,0.f,0.f,0.f,0.f,0.f,0.f};

  v4f  breg[8];                                  // B chunk staging registers
#if !USE_TDM
  float areg[16];                                // A chunk staging registers
#endif

  // ---------- staging helpers (macros keep everything in registers) ----------
#define LOAD_B_REGS(kc0)                                                        \
  _Pragma("unroll")                                                             \
  for (int i = 0; i < 8; ++i) {                                                 \
    int e  = (tid + i * 256) << 2;                                              \
    int kk = e >> 7;                                                            \
    int nl = e & (BLK_N - 1);                                                   \
    breg[i] = *(const v4f*)(kw + (size_t)((kc0) + kk) * NG + nblk0 + nl);       \
  }
#define STORE_B_REGS(buf)                                                       \
  _Pragma("unroll")                                                             \
  for (int i = 0; i < 8; ++i) {                                                 \
    int e  = (tid + i * 256) << 2;                                              \
    int kk = e >> 7;                                                            \
    int nl = e & (BLK_N - 1);                                                   \
    _Pragma("unroll")                                                           \
    for (int c = 0; c < 4; ++c) ldsB[(buf)][(nl + c) * BPAD + kk] = breg[i][c]; \
  }
#if USE_TDM
#define STAGE_A_ISSUE(kc0, buf)                                                 \
  if (wave == 0) {                                                              \
    int tdim1 = ((kc0) >= C_ && (r0 + BLK_M) >= ROWS) ? (ROWS - 1 - r0) : BLK_M;\
    tdm_tile_load(in + (size_t)r0 * C_ + (kc0),                                 \
                  (unsigned)(uintptr_t)(&ldsA[(buf)][0]),                       \
                  KC, tdim1, KC, BLK_M, C_, TDM_FLAGS_A);                       \
  }
#define STAGE_A_DRAIN()                                                         \
  if (wave == 0) { __builtin_amdgcn_s_wait_tensorcnt(0); }
#else
#define STAGE_A_ISSUE(kc0, buf)                                                 \
  _Pragma("unroll")                                                             \
  for (int i = 0; i < 16; ++i) {                                                \
    int e = tid + i * 256; int rw = e >> 6; int k = e & (KC - 1);               \
    size_t g = (size_t)(r0 + rw) * C_ + (kc0) + k;                              \
    areg[i] = (g < NELEM) ? in[g] : 0.f;                                        \
  }                                                                             \
  _Pragma("unroll")                                                             \
  for (int i = 0; i < 16; ++i) {                                                \
    int e = tid + i * 256; int rw = e >> 6; int k = e & (KC - 1);               \
    ldsA[(buf)][rw * APAD + k] = areg[i];                                       \
  }
#define STAGE_A_DRAIN()
#endif

  // ---------------- prologue: stage chunk 0 into buffer 0 ----------------
  STAGE_A_ISSUE(0, 0)
  LOAD_B_REGS(0)
  STORE_B_REGS(0)
  STAGE_A_DRAIN()
  __syncthreads();

  // ---------------- main pipeline over K chunks ----------------
  for (int ci = 0; ci < NCHUNK; ++ci) {
    const int  buf   = ci & 1;
    const int  kc0   = ci * KC;
    const bool more  = (ci + 1) < NCHUNK;

    if (more) {
      STAGE_A_ISSUE(kc0 + KC, buf ^ 1)           // TDM for next chunk (async)
      LOAD_B_REGS(kc0 + KC)                      // global loads in flight
    }

    // compute current chunk: pure LDS + WMMA
    const float amul = ((kc0 >= C_) && lastT) ? 0.f : 1.f;
    const float* aF = &ldsA[buf][(waveM * 16 + halfrow) * APAD + sel2];
    const float* bF = &ldsB[buf][(nwave + halfrow) * BPAD + sel2];
    #pragma unroll
    for (int ks = 0; ks < KC / 4; ++ks) {
      v2f a = *(const v2f*)(aF + ks * 4);
      a.x *= amul; a.y *= amul;
      #pragma unroll
      for (int nt = 0; nt < 4; ++nt) {
        v2f b = *(const v2f*)(bF + nt * 16 * BPAD + ks * 4);
        acc[nt] = __builtin_amdgcn_wmma_f32_16x16x4_f32(
            false, a, false, b, (short)0, acc[nt], false, false);
      }
    }

    if (more) {
      STORE_B_REGS(buf ^ 1)                      // drains the global loads
      STAGE_A_DRAIN()                            // s_wait_tensorcnt 0 (wave 0)
    }
    __syncthreads();
  }

  // ---------------- epilogue: bias + activation, scatter by gate ----------------
  const int rowoff = (lane >> 4) * 8;            // C/D layout: upper lanes are M+8
  #pragma unroll
  for (int nt = 0; nt < 4; ++nt) {
    const int col = nblk0 + nwave + nt * 16 + halfrow;
    const int g   = col >> 9;                    // 0=x, 1=f, 2=o
    const int u   = col & (U_ - 1);
    const float bv = bias[col];
    float* dst = (g == 0) ? xact : ((g == 1) ? fgate : ogate);
    #pragma unroll
    for (int v = 0; v < 8; ++v) {
      const int row = m0 + v + rowoff;
      float val = acc[nt][v] + bv;
      val = (g == 0) ? tanhf(val) : (1.f / (1.f + __expf(-val)));
      dst[(size_t)row * U_ + u] = val;
    }
  }
#undef LOAD_B_REGS
#undef STORE_B_REGS
#undef STAGE_A_ISSUE
#undef STAGE_A_DRAIN
}

// c_t = f_t * c_{t-1} + (1 - f_t) * x_t ;  h_t = o_t * c_t
__global__ __launch_bounds__(256) void qrnn_scan(
    const float* __restrict__ xact, const float* __restrict__ fgate,
    const float* __restrict__ ogate, float* __restrict__ h)
{
  const int idx = blockIdx.x * blockDim.x + threadIdx.x;  // b*U + u
  const int b   = idx >> 9;
  const int u   = idx & (U_ - 1);
  const size_t base = (size_t)b * T_ * U_ + u;

  float c = 0.f;
  #pragma unroll 4
  for (int t = 0; t < T_; ++t) {
    const size_t off = base + (size_t)t * U_;
    if (t + 8 < T_) {                            // global_prefetch_b8 ahead of stream
      __builtin_prefetch(fgate + off + 8 * U_, 0, 3);
      __builtin_prefetch(xact  + off + 8 * U_, 0, 3);
    }
    const float f = fgate[off];
    const float x = xact[off];
    c = f * c + (1.f - f) * x;
    h[off] = ogate[off] * c;
  }
}

extern "C" void kernel_launch(void* const* d_in, const int* in_sizes, int n_in,
                              void* d_out, int out_size, void* d_ws, size_t ws_size,
                              hipStream_t stream)
{
  const float* in   = (const float*)d_in[0];   // [8,4096,512] f32
  const float* kw   = (const float*)d_in[1];   // [2,512,1536] f32 == [1024,1536] row-major
  const float* bias = (const float*)d_in[2];   // [1536] f32
  float* h = (float*)d_out;                    // [8,4096,512] f32

  float* xact  = (float*)d_ws;                 // 3 x 64 MB workspace
  float* fgate = xact  + (size_t)ROWS * U_;
  float* ogate = fgate + (size_t)ROWS * U_;

  dim3 grid(NG / BLK_N, ROWS / BLK_M);         // (12, 512)
  qrnn_gemm_gates<<<grid, 256, 0, stream>>>(in, kw, bias, xact, fgate, ogate);
  qrnn_scan<<<(B_ * U_) / 256, 256, 0, stream>>>(xact, fgate, ogate, h);
}